// DilateAttention_6880537608923
// MI455X (gfx1250) — compile-verified
//
#include <hip/hip_runtime.h>
#include <math.h>

// DilateAttention: q,k,v f32 [8,192,56,56], head_dim=32, 6 heads, 3x3 kernel dil=2.
// Memory-bound (AI ~2.2 F/B). CDNA5 async global->LDS data mover transposes
// k/v into [x][chan] while staging; compute is pure ds_load_b128 + FMA.
// Row-paired blocks: rows {y0, y0+2} share staged planes (dilation = 2).

#define B_    8
#define NH    6
#define HD    32
#define CH    (NH*HD)      // 192
#define HH    56
#define WW    56
#define HW    (HH*WW)      // 3136
#define DIL   2
#define SCALE 0.17677669529663687f  // 32^-0.5

#define PITCH   36                // dwords per pixel record (32 ch + pad): 144B, 16B-aligned
#define PLANE   (WW*PITCH)        // 2016 dwords per staged row-plane
#define NPLANE  4                 // planes y0-2, y0, y0+2, y0+4
#define STAGE_F (NPLANE*PLANE)    // 8064 floats = 32256 B per tensor

__device__ __forceinline__ unsigned lds_off_u32(const void* p) {
  // flat LDS address: aperture in [63:32], byte offset in [31:0]
  return (unsigned)(unsigned long long)p;
}

__device__ __forceinline__ void async_b32(unsigned lds_byte,
                                          const float* gbase,
                                          unsigned byte_off) {
  // per-lane 4B global -> LDS (transposing scatter), tracked by ASYNCcnt
  asm volatile("global_load_async_to_lds_b32 %0, %1, %2"
               :
               : "v"(lds_byte), "v"(byte_off), "s"(gbase)
               : "memory");
}

__device__ __forceinline__ void wait_async0() {
  asm volatile("s_wait_asynccnt 0" ::: "memory");
}

__global__ __launch_bounds__(128)
void dilate_attn_kernel(const float* __restrict__ q,
                        const float* __restrict__ k,
                        const float* __restrict__ v,
                        float* __restrict__ out) {
  __shared__ __align__(16) float sK[STAGE_F];   // [plane][x][c] pitch 36
  __shared__ __align__(16) float sV[STAGE_F];

  const int tid = threadIdx.x;
  const int lx  = tid & 63;       // pixel x (active for lx < 56)
  // tid>>6 is identical for all 32 lanes of a wave; readfirstlane makes the
  // compiler treat it (and everything derived) as an SGPR value.
  const int hu  = __builtin_amdgcn_readfirstlane(tid >> 6);

  // pair decode: p in [0,28), y0 = 4*(p>>1) + (p&1); rows {y0, y0+2} cover all 56
  const int bidx = blockIdx.x;
  const int p    = bidx % 28;
  const int th   = bidx / 28;
  const int head = th % NH;
  const int b    = th / NH;
  const int y0   = 4 * (p >> 1) + (p & 1);
  const int chan0 = head * HD;

  const unsigned sK0 = lds_off_u32(sK);
  const unsigned sV0 = lds_off_u32(sV);

  // ---- async-stage 4 planes (yy = y0-2+2*pr) x 32 ch x 56 px, transposed ----
  // waves 0-1 stage K, waves 2-3 stage V; line math is scalar (SGPR base adds),
  // the per-lane part is a loop-invariant x offset.
  {
    const float*  gsrc  = hu ? v : k;           // s_cselect (uniform)
    const unsigned lds0 = hu ? sV0 : sK0;       // s_cselect (uniform)
    const unsigned gxoff = (unsigned)(lx * 4);            // per-lane, invariant
    const unsigned lxoff = (unsigned)(lx * PITCH * 4);    // per-lane, invariant
    if (lx < WW) {
#pragma unroll
      for (int pr = 0; pr < NPLANE; ++pr) {
        const int yy = y0 - 2 + 2 * pr;
        if (yy >= 0 && yy < HH) {                          // scalar branch
          const size_t gline0 = (size_t)(((b * CH + chan0) * HH + yy) * WW) * 4u;
          const unsigned lline = lds0 + (unsigned)(pr * PLANE * 4) + lxoff;
#pragma unroll
          for (int c = 0; c < HD; ++c) {
            // scalar side: tensor base + line offset (s_add); vector side: x*4
            const float* gp = (const float*)((const char*)gsrc + gline0 +
                                             (size_t)c * (HW * 4u));
            async_b32(lline + (unsigned)(c * 4), gp, gxoff);
          }
        }
      }
    }
  }

  // ---- overlap: pull q into registers while the DMA runs (coalesced: lane = x)
  const int y = y0 + 2 * hu;   // this wave's output row (scalar)
  float qv[HD];
  if (lx < WW) {
    const float* qp = q + (size_t)(b * CH + chan0) * HW + y * WW + lx;
#pragma unroll
    for (int c = 0; c < HD; ++c) qv[c] = qp[(size_t)c * HW];
  }

  wait_async0();     // wave-uniform
  __syncthreads();

  if (lx < WW) {
    const int x = lx;

    // ---- scores: 9 neighbor dots over 32 dims via 8x ds_load_b128 each ----
    // row-half hu uses planes {hu, hu+1, hu+2}: pr = hu + r (scalar)
    float sc[9];
#pragma unroll
    for (int n = 0; n < 9; ++n) {
      const int r  = n / 3;
      const int dx = (n % 3 - 1) * DIL;
      const int xx = x + dx;
      const int yy = y + (r - 1) * DIL;   // scalar
      float s = 0.f;
      if (yy >= 0 && yy < HH && xx >= 0 && xx < WW) {
        const float4* kp = (const float4*)(sK + (hu + r) * PLANE + xx * PITCH);
        float4 acc = make_float4(0.f, 0.f, 0.f, 0.f);
#pragma unroll
        for (int u = 0; u < HD / 4; ++u) {
          const float4 kv = kp[u];
          acc.x = fmaf(qv[4 * u + 0], kv.x, acc.x);
          acc.y = fmaf(qv[4 * u + 1], kv.y, acc.y);
          acc.z = fmaf(qv[4 * u + 2], kv.z, acc.z);
          acc.w = fmaf(qv[4 * u + 3], kv.w, acc.w);
        }
        s = ((acc.x + acc.y) + (acc.z + acc.w)) * SCALE;
      }
      sc[n] = s;  // zero-padded border: score exactly 0 (stays in softmax)
    }

    // ---- softmax over 9 ----
    float m = sc[0];
#pragma unroll
    for (int n = 1; n < 9; ++n) m = fmaxf(m, sc[n]);
    float e[9];
    float sum = 0.f;
#pragma unroll
    for (int n = 0; n < 9; ++n) {
      e[n] = __expf(sc[n] - m);
      sum += e[n];
    }
    const float inv = 1.f / sum;

    // ---- weighted sum of neighbor values (float4 accumulators) ----
    float4 o4[HD / 4];
#pragma unroll
    for (int u = 0; u < HD / 4; ++u) o4[u] = make_float4(0.f, 0.f, 0.f, 0.f);
#pragma unroll
    for (int n = 0; n < 9; ++n) {
      const int r  = n / 3;
      const int dx = (n % 3 - 1) * DIL;
      const int xx = x + dx;
      const int yy = y + (r - 1) * DIL;   // scalar
      if (yy >= 0 && yy < HH && xx >= 0 && xx < WW) {
        const float w = e[n] * inv;
        const float4* vp = (const float4*)(sV + (hu + r) * PLANE + xx * PITCH);
#pragma unroll
        for (int u = 0; u < HD / 4; ++u) {
          const float4 vv = vp[u];
          o4[u].x = fmaf(w, vv.x, o4[u].x);
          o4[u].y = fmaf(w, vv.y, o4[u].y);
          o4[u].z = fmaf(w, vv.z, o4[u].z);
          o4[u].w = fmaf(w, vv.w, o4[u].w);
        }
      }
    }

    // ---- out[b, y, x, head*32 + c]: one full 128B line per pixel per block ----
    float4* op = (float4*)(out + (((size_t)(b * HH + y) * WW + x) * CH + chan0));
#pragma unroll
    for (int u = 0; u < HD / 4; ++u) op[u] = o4[u];
  }
}

extern "C" void kernel_launch(void* const* d_in, const int* in_sizes, int n_in,
                              void* d_out, int out_size, void* d_ws, size_t ws_size,
                              hipStream_t stream) {
  const float* q = (const float*)d_in[0];
  const float* k = (const float*)d_in[1];
  const float* v = (const float*)d_in[2];
  float* out = (float*)d_out;
  dim3 grid(B_ * NH * 28);  // 1344 workgroups: (b, head, row-pair {y0, y0+2})
  dim3 block(128);          // 4 waves (wave32)
  hipLaunchKernelGGL(dilate_attn_kernel, grid, block, 0, stream, q, k, v, out);
}